// ContinuousSSM_83159156785382
// MI455X (gfx1250) — compile-verified
//
#include <hip/hip_runtime.h>
#include <math.h>

// MI455X / gfx1250, wave32. All GEMMs use v_wmma_f32_16x16x4_f32 (full f32
// precision to match the reference). Each wave computes a 16x64 output strip
// (4 WMMA accumulators share one A fragment). The K-loop is explicitly
// software-pipelined (double-buffered fragment groups in distinct registers)
// so global loads for group k+4 overlap the 4 chained v_wmma of group k.
// SSM fixed-point loop collapsed to step detection + one exact replay pass.

typedef __attribute__((ext_vector_type(2))) float v2f;
typedef __attribute__((ext_vector_type(8))) float v8f;

#define D_MODEL   512
#define D_INNER   1024
#define D_STATE   64
#define D_CONV    4
#define LSEQ      256
#define MAX_STEPS 10
#define THRESH    1e-4f
#define DT_BASE   0.1f

// ---------------------------------------------------------------- LayerNorm
template<int DW>
__global__ void ln_kernel(const float* __restrict__ in, const float* __restrict__ g,
                          const float* __restrict__ bta, const float* __restrict__ resid,
                          float* __restrict__ out)
{
    __shared__ float red[256];
    int l = blockIdx.x, t = threadIdx.x;
    const float* row = in + (size_t)l * DW;
    float s = 0.f;
    for (int i = t; i < DW; i += 256) s += row[i];
    red[t] = s; __syncthreads();
    for (int w = 128; w > 0; w >>= 1) { if (t < w) red[t] += red[t + w]; __syncthreads(); }
    float mean = red[0] / (float)DW;
    __syncthreads();
    float s2 = 0.f;
    for (int i = t; i < DW; i += 256) { float d = row[i] - mean; s2 += d * d; }
    red[t] = s2; __syncthreads();
    for (int w = 128; w > 0; w >>= 1) { if (t < w) red[t] += red[t + w]; __syncthreads(); }
    float rstd = 1.0f / sqrtf(red[0] / (float)DW + 1e-5f);
    for (int i = t; i < DW; i += 256) {
        float v = (row[i] - mean) * rstd * g[i] + bta[i];
        if (resid) v += resid[(size_t)l * DW + i];
        out[(size_t)l * DW + i] = v;
    }
}

// ------------------------------------------------------- f32 WMMA GEMM
// C[M,N] = A[M,K] @ B[K,N], row-major. One wave per 16x64 strip. K stepped by
// 4, pipelined 2 deep (requires K % 8 == 0 -- true for all uses here).
// MODE 0: none | MODE 1: +bias, exact GELU | MODE 2: +bias, softplus * DT_BASE
template<int MODE>
__global__ void gemm_wmma(const float* __restrict__ A, const float* __restrict__ B,
                          const float* __restrict__ bias, float* __restrict__ C,
                          int M, int N, int K)
{
    int wid  = threadIdx.x >> 5;
    int lane = threadIdx.x & 31;
    int nstrips = N >> 6;                       // strips of 64 columns
    int strip = blockIdx.x * (blockDim.x >> 5) + wid;
    if (strip >= (M >> 4) * nstrips) return;    // whole wave exits: EXEC stays all-1s
    int sm = strip / nstrips, sn = strip % nstrips;
    int half = lane >> 4, t = lane & 15;

    const float* Ap = A + (size_t)(sm * 16 + t) * K + 2 * half;   // M=t, K base +2*half
    const float* Bp = B + (size_t)(2 * half) * N + sn * 64 + t;   // K base +2*half, N cols

#define LOADF(A_, B0_, B1_, B2_, B3_)                                   \
    { A_.x = Ap[0]; A_.y = Ap[1];                                       \
      const float* br0 = Bp; const float* br1 = Bp + N;                 \
      B0_.x = br0[0];  B0_.y = br1[0];                                  \
      B1_.x = br0[16]; B1_.y = br1[16];                                 \
      B2_.x = br0[32]; B2_.y = br1[32];                                 \
      B3_.x = br0[48]; B3_.y = br1[48];                                 \
      Ap += 4; Bp += (size_t)4 * N; }

#define WMMA4(A_, B0_, B1_, B2_, B3_)                                                                      \
    { acc0 = __builtin_amdgcn_wmma_f32_16x16x4_f32(false, A_, false, B0_, (short)0, acc0, false, false);   \
      acc1 = __builtin_amdgcn_wmma_f32_16x16x4_f32(false, A_, false, B1_, (short)0, acc1, false, false);   \
      acc2 = __builtin_amdgcn_wmma_f32_16x16x4_f32(false, A_, false, B2_, (short)0, acc2, false, false);   \
      acc3 = __builtin_amdgcn_wmma_f32_16x16x4_f32(false, A_, false, B3_, (short)0, acc3, false, false); }

    v8f acc0 = {}, acc1 = {}, acc2 = {}, acc3 = {};
    v2f a0, b00, b01, b02, b03;
    v2f a1, b10, b11, b12, b13;

    LOADF(a0, b00, b01, b02, b03);                      // group k=0
    for (int k = 0; k < K; k += 8) {
        LOADF(a1, b10, b11, b12, b13);                  // group k+4 (overlaps WMMA4 below)
        WMMA4(a0, b00, b01, b02, b03);
        if (k + 8 < K) LOADF(a0, b00, b01, b02, b03);   // group k+8
        WMMA4(a1, b10, b11, b12, b13);
    }
#undef LOADF
#undef WMMA4

    v8f accs[4] = {acc0, acc1, acc2, acc3};
    #pragma unroll
    for (int j = 0; j < 4; ++j) {
        int col = sn * 64 + 16 * j + t;
        float bv = (MODE != 0) ? bias[col] : 0.0f;
        #pragma unroll
        for (int r = 0; r < 8; ++r) {
            float v = accs[j][r];
            if (MODE == 1) { v += bv; v = 0.5f * v * (1.0f + erff(v * 0.70710678118654752f)); }
            if (MODE == 2) { v += bv; v = (v > 20.0f ? v : log1pf(expf(v))) * DT_BASE; }
            int row = sm * 16 + r + 8 * half;   // C/D layout: lanes 16-31 hold M = r+8
            C[(size_t)row * N + col] = v;
        }
    }
}

// --------------------------------------------- depthwise causal conv + SiLU
__global__ void conv_silu_kernel(const float* __restrict__ xz, const float* __restrict__ cw,
                                 const float* __restrict__ cb, float* __restrict__ act)
{
    int idx = blockIdx.x * blockDim.x + threadIdx.x;
    if (idx >= LSEQ * D_INNER) return;
    int l = idx >> 10, d = idx & (D_INNER - 1);
    float s = cb[d];
    #pragma unroll
    for (int j = 0; j < D_CONV; ++j) {
        int ls = l + j - (D_CONV - 1);
        if (ls >= 0) s += cw[d * D_CONV + j] * xz[(size_t)ls * (2 * D_INNER) + d];
    }
    act[idx] = s / (1.0f + expf(-s));   // SiLU
}

__global__ void neg_exp_kernel(const float* __restrict__ in, float* __restrict__ out, int n)
{
    int i = blockIdx.x * blockDim.x + threadIdx.x;
    if (i < n) out[i] = -expf(in[i]);
}

// --------------------------------- per-position per-step ||dh||^2 sums S[l][k]
__global__ void ssm_diff_kernel(const float* __restrict__ dt, const float* __restrict__ act,
                                const float* __restrict__ Bm, const float* __restrict__ Amat,
                                float* __restrict__ S)
{
    __shared__ float red[256];
    __shared__ float Bsh[D_STATE];
    int l = blockIdx.x, t = threadIdx.x;
    if (t < D_STATE) Bsh[t] = Bm[l * D_STATE + t];
    __syncthreads();

    float accs[MAX_STEPS];
    #pragma unroll
    for (int k = 0; k < MAX_STEPS; ++k) accs[k] = 0.f;

    for (int i = t; i < D_INNER * D_STATE; i += 256) {
        int d = i >> 6, n = i & 63;
        float dtv = dt[(size_t)l * D_INNER + d];
        float bx  = dtv * act[(size_t)l * D_INNER + d] * Bsh[n];
        float a   = expf(dtv * Amat[i]);
        float h = 0.f;
        #pragma unroll
        for (int k = 0; k < MAX_STEPS; ++k) {
            float hn = fmaf(a, h, bx);
            float dh = hn - h;
            accs[k] = fmaf(dh, dh, accs[k]);
            h = 0.5f * (hn + h);
        }
    }
    for (int k = 0; k < MAX_STEPS; ++k) {
        red[t] = accs[k]; __syncthreads();
        for (int w = 128; w > 0; w >>= 1) { if (t < w) red[t] += red[t + w]; __syncthreads(); }
        if (t == 0) S[l * MAX_STEPS + k] = red[0];
        __syncthreads();
    }
}

// ------------- pick k* = first step whose global diff < THRESH (else last step)
__global__ void kstar_kernel(const float* __restrict__ S, int* __restrict__ kstar)
{
    __shared__ float red[256];
    __shared__ float diffs[MAX_STEPS];
    int t = threadIdx.x;           // t == position l
    for (int k = 0; k < MAX_STEPS; ++k) {
        red[t] = sqrtf(S[t * MAX_STEPS + k]); __syncthreads();
        for (int w = 128; w > 0; w >>= 1) { if (t < w) red[t] += red[t + w]; __syncthreads(); }
        if (t == 0) diffs[k] = red[0] / (float)LSEQ;
        __syncthreads();
    }
    if (t == 0) {
        int ks = MAX_STEPS - 1;
        for (int k = 0; k < MAX_STEPS; ++k) { if (diffs[k] < THRESH) { ks = k; break; } }
        kstar[0] = ks;
    }
}

// ------------- exact replay to h_new at step k*, contract with Cm, gate with z
__global__ void ssm_y_kernel(const float* __restrict__ dt, const float* __restrict__ act,
                             const float* __restrict__ Bm, const float* __restrict__ Cm,
                             const float* __restrict__ Amat, const float* __restrict__ Dvec,
                             const float* __restrict__ xz, const int* __restrict__ kstar,
                             float* __restrict__ y)
{
    int idx = blockIdx.x * blockDim.x + threadIdx.x;
    if (idx >= LSEQ * D_INNER) return;
    int l = idx >> 10, d = idx & (D_INNER - 1);
    int ks = kstar[0];
    float dtv = dt[idx], xv = act[idx];
    float dtx = dtv * xv;
    const float* Arow = Amat + d * D_STATE;
    const float* Brow = Bm + l * D_STATE;
    const float* Crow = Cm + l * D_STATE;
    float acc = 0.f;
    for (int n = 0; n < D_STATE; ++n) {
        float a  = expf(dtv * Arow[n]);
        float bx = dtx * Brow[n];
        float h = 0.f, hn = bx;
        for (int k = 0; k <= ks; ++k) {
            hn = fmaf(a, h, bx);
            if (k < ks) h = 0.5f * (hn + h);
        }
        acc = fmaf(hn, Crow[n], acc);
    }
    float yv = acc + Dvec[d] * xv;
    float z = xz[(size_t)l * (2 * D_INNER) + D_INNER + d];
    yv *= z / (1.0f + expf(-z));    // * silu(z)
    y[idx] = yv;
}

// ---------------------------------------------------------------- launcher
extern "C" void kernel_launch(void* const* d_in, const int* in_sizes, int n_in,
                              void* d_out, int out_size, void* d_ws, size_t ws_size,
                              hipStream_t stream)
{
    const float* x       = (const float*)d_in[0];
    const float* W_in    = (const float*)d_in[1];
    const float* conv_w  = (const float*)d_in[2];
    const float* conv_b  = (const float*)d_in[3];
    const float* A_log   = (const float*)d_in[4];
    const float* W_B     = (const float*)d_in[5];
    const float* W_C     = (const float*)d_in[6];
    const float* Dv      = (const float*)d_in[7];
    const float* dt_w1   = (const float*)d_in[8];
    const float* dt_b1   = (const float*)d_in[9];
    const float* dt_w2   = (const float*)d_in[10];
    const float* dt_b2   = (const float*)d_in[11];
    const float* W_out   = (const float*)d_in[12];
    const float* ln_in_g = (const float*)d_in[13];
    const float* ln_in_b = (const float*)d_in[14];
    const float* ln_out_g= (const float*)d_in[15];
    const float* ln_out_b= (const float*)d_in[16];

    float* ws   = (float*)d_ws;
    float* xn   = ws;                 // 256*512
    float* xz   = xn  + 131072;       // 256*2048
    float* act  = xz  + 524288;       // 256*1024
    float* t1g  = act + 262144;       // 256*256
    float* dtb  = t1g + 65536;        // 256*1024
    float* Bm   = dtb + 262144;       // 256*64
    float* Cm   = Bm  + 16384;        // 256*64
    float* Am   = Cm  + 16384;        // 1024*64
    float* S    = Am  + 65536;        // 256*10
    float* yb   = S   + 2560;         // 256*1024
    float* ypre = yb  + 262144;       // 256*512
    int*   kst  = (int*)(ypre + 131072);

    // 1) input LayerNorm
    ln_kernel<D_MODEL><<<LSEQ, 256, 0, stream>>>(x, ln_in_g, ln_in_b, nullptr, xn);
    // 2) A = -exp(A_log)
    neg_exp_kernel<<<(D_INNER * D_STATE + 255) / 256, 256, 0, stream>>>(A_log, Am, D_INNER * D_STATE);
    // 3) xz = xn @ W_in   (256 x 2048, K=512)
    { int strips = (LSEQ/16) * (2*D_INNER/64);
      gemm_wmma<0><<<(strips + 3)/4, 128, 0, stream>>>(xn, W_in, nullptr, xz, LSEQ, 2*D_INNER, D_MODEL); }
    // 4) depthwise causal conv + SiLU -> act
    conv_silu_kernel<<<(LSEQ*D_INNER)/256, 256, 0, stream>>>(xz, conv_w, conv_b, act);
    // 5) t1g = gelu(act @ dt_w1 + b1)   (256 x 256, K=1024)
    { int strips = (LSEQ/16) * (D_INNER/4/64);
      gemm_wmma<1><<<(strips + 3)/4, 128, 0, stream>>>(act, dt_w1, dt_b1, t1g, LSEQ, D_INNER/4, D_INNER); }
    // 6) dt = softplus(t1g @ dt_w2 + b2) * DT_BASE   (256 x 1024, K=256)
    { int strips = (LSEQ/16) * (D_INNER/64);
      gemm_wmma<2><<<(strips + 3)/4, 128, 0, stream>>>(t1g, dt_w2, dt_b2, dtb, LSEQ, D_INNER, D_INNER/4); }
    // 7) Bm, Cm   (256 x 64, K=1024)
    { int strips = (LSEQ/16) * (D_STATE/64);
      gemm_wmma<0><<<(strips + 3)/4, 128, 0, stream>>>(act, W_B, nullptr, Bm, LSEQ, D_STATE, D_INNER);
      gemm_wmma<0><<<(strips + 3)/4, 128, 0, stream>>>(act, W_C, nullptr, Cm, LSEQ, D_STATE, D_INNER); }
    // 8) per-step global diff sums, then k*
    ssm_diff_kernel<<<LSEQ, 256, 0, stream>>>(dtb, act, Bm, Am, S);
    kstar_kernel<<<1, 256, 0, stream>>>(S, kst);
    // 9) exact SSM output + D skip + SiLU(z) gating
    ssm_y_kernel<<<(LSEQ*D_INNER)/256, 256, 0, stream>>>(dtb, act, Bm, Cm, Am, Dv, xz, kst, yb);
    // 10) ypre = y @ W_out   (256 x 512, K=1024)
    { int strips = (LSEQ/16) * (D_MODEL/64);
      gemm_wmma<0><<<(strips + 3)/4, 128, 0, stream>>>(yb, W_out, nullptr, ypre, LSEQ, D_MODEL, D_INNER); }
    // 11) output LayerNorm + residual
    ln_kernel<D_MODEL><<<LSEQ, 256, 0, stream>>>(ypre, ln_out_g, ln_out_b, x, (float*)d_out);
}